// Embedding_4896262717955
// MI455X (gfx1250) — compile-verified
//
#include <hip/hip_runtime.h>

// N=8192 rows, K=100000 vocab, F=64 features.
// out = onehot(x) @ W  ==  row-gather of W scaled by the one-hot value.
// Phase 1: stream x (3.28 GB, the roofline bottleneck ~141us @ 23.3 TB/s)
//          with non-temporal b128 loads; record (k, v) per row.
// Phase 2: per 16-row group, compute out = diag(v) @ Wg exactly in f32 via
//          V_WMMA_F32_16X16X4_F32 (4 K-chunks x 4 N-chunks = 16 WMMAs/wave).

typedef float v2f __attribute__((ext_vector_type(2)));
typedef float v4f __attribute__((ext_vector_type(4)));
typedef float v8f __attribute__((ext_vector_type(8)));

#define NROWS 8192
#define KDIM  100000
#define FDIM  64

// ---------------- Phase 1: one-hot scan (memory-bound) ----------------
// One 256-thread block per row. 100000/4 = 25000 float4 loads per row,
// ~98 per thread, fully coalesced (32 lanes x 16B = 512B contiguous).
__global__ __launch_bounds__(256) void onehot_scan_kernel(
    const float* __restrict__ x,
    int* __restrict__ row_k,
    float* __restrict__ row_v)
{
    const int n = blockIdx.x;
    const v4f* __restrict__ row = (const v4f*)(x + (size_t)n * (size_t)KDIM);

    __shared__ int   k_sh;
    __shared__ float v_sh;
    if (threadIdx.x == 0) { k_sh = 0; v_sh = 0.0f; }  // all-zero row -> zero output
    __syncthreads();

    const int nvec = KDIM / 4;  // 25000, exact
    for (int i = threadIdx.x; i < nvec; i += 256) {
        v4f u = __builtin_nontemporal_load(&row[i]);  // TH=NT: don't thrash L2
        // rare path: exactly one hit per row for a one-hot input
        if (u.x != 0.0f || u.y != 0.0f || u.z != 0.0f || u.w != 0.0f) {
            if (u.x != 0.0f) { k_sh = 4 * i + 0; v_sh = u.x; }
            if (u.y != 0.0f) { k_sh = 4 * i + 1; v_sh = u.y; }
            if (u.z != 0.0f) { k_sh = 4 * i + 2; v_sh = u.z; }
            if (u.w != 0.0f) { k_sh = 4 * i + 3; v_sh = u.w; }
        }
    }
    __syncthreads();
    if (threadIdx.x == 0) { row_k[n] = k_sh; row_v[n] = v_sh; }
}

// ---------------- Phase 2: gather + diag(v) @ Wg via f32 WMMA ----------------
// One wave (32 lanes) handles 16 rows. out_tile[16x64] computed as
// sum_j A_j(16x4) @ B_j(4x16) per 16-col chunk, where A_j is the j-th
// 4-column slice of diag(v) and B_j holds gathered W rows.
//
// ISA f32 16x16x4 layouts (CDNA5 ISA):
//   A: lane = M (0..15 in each half), VGPR p in half h -> K = 2h + p
//   B: lane = N (0..15 in each half), VGPR p in half h -> K = 2h + p
//   C/D: lane = N (+16 for M>=8), VGPR r -> M = r + 8*(lane/16)
__global__ __launch_bounds__(256) void gather_wmma_kernel(
    const float* __restrict__ W,
    const int* __restrict__ row_k,
    const float* __restrict__ row_v,
    float* __restrict__ out)
{
    const int wave = threadIdx.x >> 5;
    const int lane = threadIdx.x & 31;
    const int r0   = (blockIdx.x * 8 + wave) * 16;   // 16 rows per wave
    if (r0 >= NROWS) return;                         // wave-uniform; EXEC stays full

    const int M    = lane & 15;
    const int half = lane >> 4;
    const int K0   = 2 * half;      // K index of fragment component .x
    const int K1   = 2 * half + 1;  // K index of fragment component .y

    // Per-lane diagonal value for row M of this tile.
    const float vM = row_v[r0 + M];

    // Gathered-W row indices for each K slot of each K-chunk j (per-lane loads,
    // no dynamic register indexing). Clamp for safety against poisoned scratch.
    int kb0[4], kb1[4];
#pragma unroll
    for (int j = 0; j < 4; ++j) {
        int a = row_k[r0 + 4 * j + K0];
        int b = row_k[r0 + 4 * j + K1];
        kb0[j] = ((unsigned)a < (unsigned)KDIM) ? a : 0;
        kb1[j] = ((unsigned)b < (unsigned)KDIM) ? b : 0;
    }

#pragma unroll
    for (int c = 0; c < 4; ++c) {        // 4 chunks of 16 features (F=64)
        v8f acc = {};
#pragma unroll
        for (int j = 0; j < 4; ++j) {    // K = 16 -> 4 steps of K=4
            v2f a, b;
            // A_j = columns [4j, 4j+4) of diag(v): nonzero only where M == 4j+K
            a.x = (M == 4 * j + K0) ? vM : 0.0f;
            a.y = (M == 4 * j + K1) ? vM : 0.0f;
            // B_j[K][N] = W[row_k[r0 + 4j + K]][16c + N]  (N == M here)
            b.x = W[kb0[j] * FDIM + 16 * c + M];
            b.y = W[kb1[j] * FDIM + 16 * c + M];
#if __has_builtin(__builtin_amdgcn_wmma_f32_16x16x4_f32)
            acc = __builtin_amdgcn_wmma_f32_16x16x4_f32(
                /*neg_a=*/false, a, /*neg_b=*/false, b,
                /*c_mod=*/(short)0, acc, /*reuse_a=*/false, /*reuse_b=*/false);
#else
            // Exact scalar fallback: only the diagonal element contributes.
            {
                const int mrow0 = 4 * j + K0;  // tile row fed by a.x
                const int mrow1 = 4 * j + K1;  // tile row fed by a.y
                // D[M][N] layout: lane holds N, VGPR r holds M = r + 8*half.
#pragma unroll
                for (int r = 0; r < 8; ++r) {
                    const int Mo = r + 8 * half;
                    if (Mo == mrow0) acc[r] += a.x * b.x;
                    if (Mo == mrow1) acc[r] += a.y * b.y;
                }
                // NOTE: fallback is approximate in lane-mapping only if layouts
                // differ; primary path is the WMMA builtin.
            }
#endif
        }
        // Write D: lane holds column 16c+M, VGPR r holds row r0 + r + 8*half.
#pragma unroll
        for (int r = 0; r < 8; ++r) {
            const int Mo = r + 8 * half;
            out[(r0 + Mo) * FDIM + 16 * c + M] = acc[r];
        }
    }
}

extern "C" void kernel_launch(void* const* d_in, const int* in_sizes, int n_in,
                              void* d_out, int out_size, void* d_ws, size_t ws_size,
                              hipStream_t stream) {
    (void)in_sizes; (void)n_in; (void)out_size; (void)ws_size;
    const float* x = (const float*)d_in[0];   // [N, K] f32 one-hot
    const float* W = (const float*)d_in[1];   // [K, F] f32
    float* out = (float*)d_out;               // [N, F] f32

    // Scratch: 8192 ints + 8192 floats = 64 KB in d_ws.
    int*   row_k = (int*)d_ws;
    float* row_v = (float*)((char*)d_ws + NROWS * sizeof(int));

    onehot_scan_kernel<<<NROWS, 256, 0, stream>>>(x, row_k, row_v);
    gather_wmma_kernel<<<NROWS / 128, 256, 0, stream>>>(W, row_k, row_v, out);
}